// HypergraphAttention_41566693491337
// MI455X (gfx1250) — compile-verified
//
#include <hip/hip_runtime.h>
#include <math.h>

#define N_NODES   8192
#define HIDDEN    256
#define HEADS     4
#define HEAD_DIM  64
#define NUM_EDGES 4096
#define NUM_INC   65536
#define LN_EPS    1e-5f

typedef __attribute__((ext_vector_type(2))) float v2f;
typedef __attribute__((ext_vector_type(8))) float v8f;

// ---------------------------------------------------------------- fills
__global__ void hga_fill_f32(float* __restrict__ p, float v, int n) {
    int i = blockIdx.x * blockDim.x + threadIdx.x;
    int stride = gridDim.x * blockDim.x;
    for (; i < n; i += stride) p[i] = v;
}
__global__ void hga_fill_i32(int* __restrict__ p, int v, int n) {
    int i = blockIdx.x * blockDim.x + threadIdx.x;
    int stride = gridDim.x * blockDim.x;
    for (; i < n; i += stride) p[i] = v;
}

// ---------------------------------------------------------------- WMMA GEMM
// Out[r, c] = sum_k X[r, k] * W[c, k] + bias[c]   (i.e. X @ W^T + b)
// X: [rows, 256] row-major, W: [256, 256] row-major, Out: [rows, 256]
// One wave computes a 16x64 strip: one A fragment feeds 4 WMMAs per K-step
// (4x register-level A reuse), 64 K-steps -> 256 V_WMMA_F32_16X16X4_F32.
__global__ __launch_bounds__(256)
void hga_gemm_wmma(const float* __restrict__ X, const float* __restrict__ W,
                   const float* __restrict__ bias, float* __restrict__ Out) {
    const int wave = threadIdx.x >> 5;
    const int lane = threadIdx.x & 31;
    const int task = blockIdx.x * 8 + wave;       // 8 waves per block
    const int col_groups = HIDDEN / 64;           // 4 strips of 64 cols
    const int tm = task / col_groups;             // row tile (16 rows)
    const int tg = task % col_groups;             // column strip (64 cols)
    const int hi = lane >> 4;                     // 0: K pair {0,1}, 1: {2,3}
    const int l  = lane & 15;

    const float* arow = X + (size_t)(tm * 16 + l) * HIDDEN;          // A: 16 rows of X
    const float* brow = W + (size_t)(tg * 64 + l) * HIDDEN;          // B[k,n] = W[n,k]

    v8f acc0 = {0.f,0.f,0.f,0.f,0.f,0.f,0.f,0.f};
    v8f acc1 = acc0, acc2 = acc0, acc3 = acc0;

    #pragma unroll 4
    for (int k0 = 0; k0 < HIDDEN; k0 += 4) {
        const int k = k0 + 2 * hi;
        v2f a;  a.x  = arow[k];                    a.y  = arow[k + 1];
        v2f b0; b0.x = brow[k];                    b0.y = brow[k + 1];
        v2f b1; b1.x = brow[16 * HIDDEN + k];      b1.y = brow[16 * HIDDEN + k + 1];
        v2f b2; b2.x = brow[32 * HIDDEN + k];      b2.y = brow[32 * HIDDEN + k + 1];
        v2f b3; b3.x = brow[48 * HIDDEN + k];      b3.y = brow[48 * HIDDEN + k + 1];
        acc0 = __builtin_amdgcn_wmma_f32_16x16x4_f32(false, a, false, b0, (short)0, acc0, false, false);
        acc1 = __builtin_amdgcn_wmma_f32_16x16x4_f32(false, a, false, b1, (short)0, acc1, false, false);
        acc2 = __builtin_amdgcn_wmma_f32_16x16x4_f32(false, a, false, b2, (short)0, acc2, false, false);
        acc3 = __builtin_amdgcn_wmma_f32_16x16x4_f32(false, a, false, b3, (short)0, acc3, false, false);
    }

    const int row0 = tm * 16 + hi * 8;             // C/D layout: lanes>=16 hold M+8
    #pragma unroll
    for (int c = 0; c < 4; ++c) {
        const int col = tg * 64 + c * 16 + l;
        const float bv = bias[col];
        const v8f acc = (c == 0) ? acc0 : (c == 1) ? acc1 : (c == 2) ? acc2 : acc3;
        #pragma unroll
        for (int v = 0; v < 8; ++v) {
            Out[(size_t)(row0 + v) * HIDDEN + col] = acc[v] + bv;
        }
    }
}

// ---------------------------------------------------------------- logits / segmax
__device__ inline void hga_atomic_max_f32(float* addr, float val) {
    unsigned int* ua = (unsigned int*)addr;
    unsigned int old = __float_as_uint(*(volatile float*)addr);
    while (__uint_as_float(old) < val) {
        unsigned int assumed = old;
        old = atomicCAS(ua, assumed, __float_as_uint(val));
        if (old == assumed) break;
    }
}

__global__ void hga_logits(const float* __restrict__ xw,
                           const int* __restrict__ node_idx,
                           const int* __restrict__ edge_idx,
                           const float* __restrict__ edge_att,
                           float* __restrict__ logits,
                           float* __restrict__ seg_max,
                           int* __restrict__ csr_cnt,
                           float* __restrict__ counts) {
    const int gid = blockIdx.x * blockDim.x + threadIdx.x;  // NUM_INC*HEADS threads
    const int i = gid >> 2;
    const int h = gid & 3;
    const int node = node_idx[i];
    const int e    = edge_idx[i];
    const float* f = xw + (size_t)node * HIDDEN + h * HEAD_DIM;
    const float* w = edge_att + h * HEAD_DIM;
    float s = 0.f;
    #pragma unroll 8
    for (int d = 0; d < HEAD_DIM; ++d) s += f[d] * w[d];
    logits[i * HEADS + h] = s;
    hga_atomic_max_f32(&seg_max[e * HEADS + h], s);
    if (h == 0) {
        atomicAdd(&csr_cnt[e], 1);
        atomicAdd(&counts[node], 1.0f);
    }
}

// ---------------------------------------------------------------- exp + segsum
__global__ void hga_exp_segsum(const int* __restrict__ edge_idx,
                               float* __restrict__ pbuf,        // in: logits, out: p
                               const float* __restrict__ seg_max,
                               float* __restrict__ seg_sum) {
    const int gid = blockIdx.x * blockDim.x + threadIdx.x;
    const int i = gid >> 2;
    const int h = gid & 3;
    const int e = edge_idx[i];
    const float p = expf(pbuf[i * HEADS + h] - seg_max[e * HEADS + h]);
    pbuf[i * HEADS + h] = p;
    atomicAdd(&seg_sum[e * HEADS + h], p);
}

// ---------------------------------------------------------------- attn_mean
__global__ void hga_attn_mean(const int* __restrict__ edge_idx,
                              const float* __restrict__ pbuf,
                              const float* __restrict__ seg_sum,
                              const int* __restrict__ csr_cnt,
                              float* __restrict__ attn_mean) {
    const int i = blockIdx.x * blockDim.x + threadIdx.x;    // NUM_INC threads
    const int e = edge_idx[i];
    float am = 0.f;
    if (csr_cnt[e] >= 2) {
        #pragma unroll
        for (int h = 0; h < HEADS; ++h)
            am += pbuf[i * HEADS + h] / seg_sum[e * HEADS + h];
        am *= (1.0f / HEADS);
    }
    attn_mean[i] = am;
}

// ---------------------------------------------------------------- scatter contrib
__global__ void hga_scatter(const float* __restrict__ x,
                            const int* __restrict__ node_idx,
                            const float* __restrict__ attn_mean,
                            float* __restrict__ out_pre) {
    const int i = blockIdx.x;                 // incidence
    const int c = threadIdx.x;                // hidden dim
    const float am = attn_mean[i];
    if (am != 0.f) {
        const int node = node_idx[i];
        atomicAdd(&out_pre[(size_t)node * HIDDEN + c],
                  am * x[(size_t)node * HIDDEN + c]);
    }
}

// ---------------------------------------------------------------- degree normalize
__global__ void hga_normalize(float* __restrict__ out_pre,
                              const float* __restrict__ counts) {
    const int idx = blockIdx.x * blockDim.x + threadIdx.x;  // N_NODES*HIDDEN
    const int n = idx >> 8;
    float c = counts[n];
    c = c < 1.f ? 1.f : c;
    out_pre[idx] /= c;
}

// ---------------------------------------------------------------- layernorm
__global__ __launch_bounds__(256)
void hga_layernorm(const float* __restrict__ in, float* __restrict__ out) {
    __shared__ float red[HIDDEN];
    const int row = blockIdx.x;
    const int tid = threadIdx.x;
    const float v = in[(size_t)row * HIDDEN + tid];
    red[tid] = v;
    __syncthreads();
    for (int s = HIDDEN / 2; s > 0; s >>= 1) {
        if (tid < s) red[tid] += red[tid + s];
        __syncthreads();
    }
    const float mean = red[0] * (1.0f / HIDDEN);
    __syncthreads();
    const float d = v - mean;
    red[tid] = d * d;
    __syncthreads();
    for (int s = HIDDEN / 2; s > 0; s >>= 1) {
        if (tid < s) red[tid] += red[tid + s];
        __syncthreads();
    }
    const float var = red[0] * (1.0f / HIDDEN);
    out[(size_t)row * HIDDEN + tid] = d * rsqrtf(var + LN_EPS);
}

// ---------------------------------------------------------------- CSR scan (1 block)
__global__ __launch_bounds__(256)
void hga_scan(const int* __restrict__ cnt, int* __restrict__ offs) {
    __shared__ int part[256];
    const int tid = threadIdx.x;
    const int base = tid * 16;                // 256 threads * 16 = 4096 edges
    int local[16];
    int s = 0;
    #pragma unroll
    for (int i = 0; i < 16; ++i) { local[i] = s; s += cnt[base + i]; }
    part[tid] = s;
    __syncthreads();
    for (int off = 1; off < 256; off <<= 1) {
        int add = (tid >= off) ? part[tid - off] : 0;
        __syncthreads();
        part[tid] += add;
        __syncthreads();
    }
    const int excl = part[tid] - s;           // exclusive prefix of this segment
    #pragma unroll
    for (int i = 0; i < 16; ++i) offs[base + i] = excl + local[i];
    if (tid == 255) offs[NUM_EDGES] = part[255];
}

// ---------------------------------------------------------------- CSR fill
__global__ void hga_csr_fill(const int* __restrict__ edge_idx,
                             const int* __restrict__ offs,
                             int* __restrict__ cursor,
                             int* __restrict__ inc_list) {
    const int i = blockIdx.x * blockDim.x + threadIdx.x;    // NUM_INC
    const int e = edge_idx[i];
    const int pos = atomicAdd(&cursor[e], 1);
    inc_list[offs[e] + pos] = i;
}

// ---------------------------------------------------------------- per-edge pairs
__global__ void hga_edge_pairs(const int* __restrict__ node_idx,
                               const int* __restrict__ csr_cnt,
                               const int* __restrict__ offs,
                               const int* __restrict__ inc_list,
                               const float* __restrict__ attn_mean,
                               float* __restrict__ AW) {
    const int e = blockIdx.x;
    const int sz = csr_cnt[e];
    if (sz < 2) return;
    const int start = offs[e];
    const int total = sz * sz;
    for (int p = threadIdx.x; p < total; p += blockDim.x) {
        const int ia = inc_list[start + p / sz];
        const int ib = inc_list[start + p % sz];
        const float am = attn_mean[ia];
        if (am != 0.f)
            atomicAdd(&AW[(size_t)node_idx[ia] * N_NODES + node_idx[ib]], am);
    }
}

__global__ void hga_zero_diag(float* __restrict__ AW) {
    const int i = blockIdx.x * blockDim.x + threadIdx.x;    // N_NODES
    AW[(size_t)i * (N_NODES + 1)] = 0.f;
}

// ================================================================ launch
extern "C" void kernel_launch(void* const* d_in, const int* in_sizes, int n_in,
                              void* d_out, int out_size, void* d_ws, size_t ws_size,
                              hipStream_t stream) {
    const float* x        = (const float*)d_in[0];   // [8192, 256]
    const int*   node_idx = (const int*)  d_in[1];   // [65536]
    const int*   edge_idx = (const int*)  d_in[2];   // [65536]
    const float* W_w      = (const float*)d_in[3];   // [256, 256]
    const float* W_b      = (const float*)d_in[4];   // [256]
    const float* edge_att = (const float*)d_in[5];   // [4, 64]
    const float* out_w    = (const float*)d_in[6];   // [256, 256]
    const float* out_b    = (const float*)d_in[7];   // [256]

    float* outp = (float*)d_out;                          // [8192, 256]
    float* AW   = outp + (size_t)N_NODES * HIDDEN;        // [8192, 8192]

    // ---- workspace layout (floats then ints) ----
    float* xw        = (float*)d_ws;                      // 2,097,152 (reused for GEMM2 out)
    float* pbuf      = xw + (size_t)N_NODES * HIDDEN;     // 262,144  (logits -> p)
    float* seg_max   = pbuf + NUM_INC * HEADS;            // 16,384
    float* seg_sum   = seg_max + NUM_EDGES * HEADS;       // 16,384
    float* attn_mean = seg_sum + NUM_EDGES * HEADS;       // 65,536
    float* out_pre   = attn_mean + NUM_INC;               // 2,097,152
    float* counts    = out_pre + (size_t)N_NODES * HIDDEN;// 8,192
    int*   csr_cnt   = (int*)(counts + N_NODES);          // 4,096
    int*   offs      = csr_cnt + NUM_EDGES;               // 4,097
    int*   cursor    = offs + NUM_EDGES + 1;              // 4,096
    int*   inc_list  = cursor + NUM_EDGES;                // 65,536

    // ---- init (required every call: graph replay does not re-poison) ----
    hga_fill_f32<<<2048, 256, 0, stream>>>((float*)d_out, 0.f, out_size);
    hga_fill_f32<<<1024, 256, 0, stream>>>(out_pre, 0.f, N_NODES * HIDDEN);
    hga_fill_f32<<<32,   256, 0, stream>>>(counts,  0.f, N_NODES);
    hga_fill_f32<<<64,   256, 0, stream>>>(seg_sum, 0.f, NUM_EDGES * HEADS);
    hga_fill_f32<<<64,   256, 0, stream>>>(seg_max, -3.402823466e38f, NUM_EDGES * HEADS);
    hga_fill_i32<<<16,   256, 0, stream>>>(csr_cnt, 0, NUM_EDGES);
    hga_fill_i32<<<16,   256, 0, stream>>>(cursor,  0, NUM_EDGES);

    // ---- GEMM1: xw = x @ W_w^T + W_b  (WMMA f32 16x16x4, 16x64 strip/wave) ----
    const int gemm_blocks = (N_NODES / 16) * (HIDDEN / 64) / 8;   // 256
    hga_gemm_wmma<<<gemm_blocks, 256, 0, stream>>>(x, W_w, W_b, xw);

    // ---- logits + segment max + edge sizes + node counts ----
    hga_logits<<<NUM_INC * HEADS / 256, 256, 0, stream>>>(
        xw, node_idx, edge_idx, edge_att, pbuf, seg_max, csr_cnt, counts);

    // ---- exp + segment sum ----
    hga_exp_segsum<<<NUM_INC * HEADS / 256, 256, 0, stream>>>(edge_idx, pbuf, seg_max, seg_sum);

    // ---- attn_mean (masked for edges with < 2 incidences) ----
    hga_attn_mean<<<NUM_INC / 256, 256, 0, stream>>>(edge_idx, pbuf, seg_sum, csr_cnt, attn_mean);

    // ---- scatter attention-weighted ORIGINAL features ----
    hga_scatter<<<NUM_INC, 256, 0, stream>>>(x, node_idx, attn_mean, out_pre);

    // ---- degree normalization ----
    hga_normalize<<<N_NODES * HIDDEN / 256, 256, 0, stream>>>(out_pre, counts);

    // ---- GEMM2: tmp = out_pre @ out_w^T + out_b  (WMMA, output into xw buffer) ----
    hga_gemm_wmma<<<gemm_blocks, 256, 0, stream>>>(out_pre, out_w, out_b, xw);

    // ---- LayerNorm -> first output ----
    hga_layernorm<<<N_NODES, 256, 0, stream>>>(xw, outp);

    // ---- sparse A @ B^T via per-edge incidence pairs ----
    hga_scan<<<1, 256, 0, stream>>>(csr_cnt, offs);
    hga_csr_fill<<<NUM_INC / 256, 256, 0, stream>>>(edge_idx, offs, cursor, inc_list);
    hga_edge_pairs<<<NUM_EDGES, 256, 0, stream>>>(node_idx, csr_cnt, offs, inc_list, attn_mean, AW);
    hga_zero_diag<<<N_NODES / 256, 256, 0, stream>>>(AW);
}